// VolumeRenderer4_56848187129875
// MI455X (gfx1250) — compile-verified
//
#include <hip/hip_runtime.h>
#include <cstdint>

// Volume-renderer transfer-function lookup, MI455X (gfx1250, wave32).
// Memory-bound: 16.7M pts x (8B in + 16B out) ~= 403 MB -> ~17us @ 23.3 TB/s.
// CDNA5 paths: global_load_async_to_lds_b32 (ASYNCcnt) stages the 1.28KB TF
// table into LDS de-interleaved (wave-uniform branching, full EXEC), then
// s_wait_asynccnt + barrier; NT load/store hints keep the 403MB stream from
// rinsing L2. WMMA is inapplicable: no matmul, compute is ~50 VALU ops/pt,
// orders of magnitude under the bandwidth floor.

typedef float v4f __attribute__((ext_vector_type(4)));

#define NB 64  // N_BINS

__device__ __forceinline__ void async_b32_to_lds(uint32_t lds_off, const float* gsrc) {
  uint64_t src = (uint64_t)(uintptr_t)gsrc;
  asm volatile("global_load_async_to_lds_b32 %0, %1, off"
               :
               : "v"(lds_off), "v"(src)
               : "memory");
}

__device__ __forceinline__ v4f shade_point(float x,
                                           const float* __restrict__ sk,
                                           const v4f* __restrict__ sc) {
  // idx = clip(searchsorted(knots, x, 'right') - 1, 0, NB-2)
  //     = largest i in [0, NB-2] with knots[i] <= x, else 0.
  int idx = 0;
#pragma unroll
  for (int s = 32; s >= 1; s >>= 1) {
    int c = idx + s;
    if (c <= NB - 2 && sk[c] <= x) idx = c;
  }
  float lo = sk[idx];
  float hi = sk[idx + 1];
  float t = (x - lo) / (hi - lo);
  t = fminf(fmaxf(t, 0.0f), 1.0f);
  v4f c0 = sc[idx];
  v4f c1 = sc[idx + 1];
  return c0 + t * (c1 - c0);  // == c0*(1-t) + c1*t
}

__global__ __launch_bounds__(256) void vr_tf_kernel(
    const float* __restrict__ in,   // [N,2] interleaved (x, g)
    const float* __restrict__ tf,   // [64,5] (knot, r, g, b, a)
    const float* __restrict__ gt,   // [2,2]  (knot, val)
    float* __restrict__ out,        // [N,4]
    long long npairs) {
  __shared__ float s_knots[NB];
  __shared__ v4f s_rgba[NB];

  // LDS byte offsets of staging destinations (AS(3) ptr value == LDS offset).
  uint32_t knots_base =
      (uint32_t)(uintptr_t)(__attribute__((address_space(3))) float*)&s_knots[0];
  uint32_t rgba_base =
      (uint32_t)(uintptr_t)(__attribute__((address_space(3))) v4f*)&s_rgba[0];

  // Stage tf_table (64 rows x 5 floats = 320 dwords) into LDS with async
  // global->LDS b32 copies, de-interleaved in flight. Branches are
  // wave-uniform (lane-range splits at multiples of 32), so EXEC is all-ones
  // on every issue. Index math is shift/mask only.
  const uint32_t tid = threadIdx.x;
  {
    // Pass A: e = tid in [0,256)
    if (tid < 64u) {
      // knots: row = tid, src = tf[row*5]
      async_b32_to_lds(knots_base + tid * 4u, tf + tid * 5u);
    } else {
      // rgba elements 0..191: row = (e-64)>>2, col = (e-64)&3
      uint32_t idx = tid - 64u;
      uint32_t row = idx >> 2;
      uint32_t col = idx & 3u;
      async_b32_to_lds(rgba_base + row * 16u + col * 4u, tf + row * 5u + 1u + col);
    }
    // Pass B: rgba elements 192..255 (rows 48..63), threads 0..63
    if (tid < 64u) {
      uint32_t idx = tid + 192u;
      uint32_t row = idx >> 2;
      uint32_t col = idx & 3u;
      async_b32_to_lds(rgba_base + row * 16u + col * 4u, tf + row * 5u + 1u + col);
    }
  }
  asm volatile("s_wait_asynccnt 0x0" ::: "memory");
  __syncthreads();

  // 2-knot grad table: mod = clamp((g-k0)/(k1-k0),0,1)*(v1-v0)+v0
  // Uniform addresses -> scalar loads. Flat layout: [k0, v0, k1, v1].
  float gk0 = gt[0];
  float gv0 = gt[1];
  float gk1 = gt[2];
  float gv1 = gt[3];
  float ginv = 1.0f / (gk1 - gk0);
  float gdv = gv1 - gv0;

  const v4f* __restrict__ in4 = (const v4f*)in;  // one v4f = 2 points
  v4f* __restrict__ out4 = (v4f*)out;            // one v4f per point

  long long t0 = (long long)blockIdx.x * blockDim.x + threadIdx.x;
  long long tstride = (long long)gridDim.x * blockDim.x;
  for (long long p = t0; p < npairs; p += tstride) {
    v4f v = __builtin_nontemporal_load(in4 + p);  // {x0, g0, x1, g1}

    v4f o0 = shade_point(v.x, s_knots, s_rgba);
    v4f o1 = shade_point(v.z, s_knots, s_rgba);

    float tg0 = fminf(fmaxf((v.y - gk0) * ginv, 0.0f), 1.0f);
    float tg1 = fminf(fmaxf((v.w - gk0) * ginv, 0.0f), 1.0f);
    o0.w *= fmaf(tg0, gdv, gv0);
    o1.w *= fmaf(tg1, gdv, gv0);

    __builtin_nontemporal_store(o0, out4 + 2 * p);
    __builtin_nontemporal_store(o1, out4 + 2 * p + 1);
  }
}

extern "C" void kernel_launch(void* const* d_in, const int* in_sizes, int n_in,
                              void* d_out, int out_size, void* d_ws, size_t ws_size,
                              hipStream_t stream) {
  const float* in = (const float*)d_in[0];  // [N,2] float32
  const float* tf = (const float*)d_in[1];  // [64,5] float32
  const float* gt = (const float*)d_in[2];  // [2,2] float32
  float* out = (float*)d_out;               // [N,4] float32

  long long npts = (long long)in_sizes[0] / 2;  // in_sizes[0] = N*2 floats
  long long npairs = npts / 2;
  if (npairs <= 0) return;

  const int threads = 256;
  long long want = (npairs + threads - 1) / threads;
  long long cap = 8192;  // grid-stride: ~4 pairs/thread, amortizes LDS staging
  int blocks = (int)(want < cap ? want : cap);
  if (blocks < 1) blocks = 1;

  vr_tf_kernel<<<blocks, threads, 0, stream>>>(in, tf, gt, out, npairs);
}